// _DeRPN_ProposalLayer_2508260901853
// MI455X (gfx1250) — compile-verified
//
#include <hip/hip_runtime.h>
#include <stdint.h>

#define B_    16
#define A_    7
#define H_    100
#define W_    100
#define K_    (H_*W_)
#define N_    (2*K_*A_)      // 140000
#define PRE_  2000
#define PAD_  2048
#define POST_ 300
#define FEAT_ 16.0f
#define TNMS  0.7f
#define TILES_ 125           // 2000/16
#define MASKW  64            // u32 words per mask row (covers 2048 cols)

typedef __attribute__((ext_vector_type(2))) float    v2f;
typedef __attribute__((ext_vector_type(8))) float    v8f;
typedef __attribute__((ext_vector_type(4))) unsigned u32x4;
typedef __attribute__((ext_vector_type(8))) int      i32x8;
typedef __attribute__((ext_vector_type(4))) int      i32x4;

__device__ __forceinline__ unsigned fkey(float s) {
  unsigned u = __float_as_uint(s);
  return (u & 0x80000000u) ? ~u : (u | 0x80000000u);   // monotonic: larger key == larger float
}

// ---------------------------------------------------------------- init ----
__global__ void k_init(unsigned* mask, unsigned* hist, int* cnts,
                       float* tks, int* tki, float* out) {
  int i = blockIdx.x * blockDim.x + threadIdx.x;
  if (i < B_*PAD_*MASKW) mask[i] = 0u;
  if (i < B_*1024)       hist[i] = 0u;
  if (i < 4*B_)          cnts[i] = 0;
  if (i < B_*PAD_)       { tks[i] = -3.0e38f; tki[i] = -1; }
  if (i < B_*POST_*5)    out[i]  = 0.0f;
}

// ---------------------------------------------------- proposal decode ----
__global__ void k_prop(const float* __restrict__ sw_, const float* __restrict__ sh_,
                       const float* __restrict__ dw_, const float* __restrict__ dh_,
                       const float* __restrict__ iminfo,
                       const float* __restrict__ anw, const float* __restrict__ anh,
                       float* __restrict__ boxes, float* __restrict__ scores) {
  int t = blockIdx.x * blockDim.x + threadIdx.x;
  if (t >= B_*K_) return;
  int b = t / K_, k = t - b*K_;
  float sx = (float)(k % W_) * FEAT_;
  float sy = (float)(k / W_) * FEAT_;
  float imh = iminfo[b*3+0] - 1.0f;
  float imw = iminfo[b*3+1] - 1.0f;

  float swv[A_], shv[A_];
  float wx1[A_], wx2[A_], hy1[A_], hy2[A_];
  float mw = -3.0e38f, mh = -3.0e38f;
  float yb0 = 0.f, yb1 = 0.f, xb0 = 0.f, xb1 = 0.f;

  #pragma unroll
  for (int a = 0; a < A_; ++a) {
    size_t cs = ((size_t)b*(2*A_) + A_ + a)*K_ + k;     // scores[:, A:]
    float s_w = sw_[cs], s_h = sh_[cs];
    swv[a] = s_w; shv[a] = s_h;

    size_t c0 = ((size_t)b*(2*A_) + 2*a)*K_ + k;        // deltas[..., a, 0/1]
    float d0w = dw_[c0], d1w = dw_[c0 + K_];
    float d0h = dh_[c0], d1h = dh_[c0 + K_];

    // width string transform
    float s0 = anw[a*2+0] + sx, s1 = anw[a*2+1] + sx;
    float ln  = s1 - s0 + 1.0f;
    float ctr = s0 + 0.5f*ln;
    float pc  = d0w*ln + ctr;
    float pl  = __expf(d1w)*ln;
    wx1[a] = pc - 0.5f*pl; wx2[a] = pc + 0.5f*pl;
    // height string transform
    s0 = anh[a*2+0] + sy; s1 = anh[a*2+1] + sy;
    ln  = s1 - s0 + 1.0f;
    ctr = s0 + 0.5f*ln;
    pc  = d0h*ln + ctr;
    pl  = __expf(d1h)*ln;
    hy1[a] = pc - 0.5f*pl; hy2[a] = pc + 0.5f*pl;

    if (s_w > mw) { mw = s_w; xb0 = wx1[a]; xb1 = wx2[a]; }   // argmax w (first max)
    if (s_h > mh) { mh = s_h; yb0 = hy1[a]; yb1 = hy2[a]; }   // argmax h
  }

  size_t base = (size_t)b*N_ + (size_t)k*(2*A_);
  #pragma unroll
  for (int a = 0; a < A_; ++a) {
    size_t nw = base + a, nh = base + A_ + a;
    float* pw = boxes + nw*4;
    pw[0] = fminf(fmaxf(wx1[a], 0.f), imw);
    pw[1] = fminf(fmaxf(yb0,    0.f), imh);
    pw[2] = fminf(fmaxf(wx2[a], 0.f), imw);
    pw[3] = fminf(fmaxf(yb1,    0.f), imh);
    scores[nw] = swv[a] * mh;
    float* ph = boxes + nh*4;
    ph[0] = fminf(fmaxf(xb0,    0.f), imw);
    ph[1] = fminf(fmaxf(hy1[a], 0.f), imh);
    ph[2] = fminf(fmaxf(xb1,    0.f), imw);
    ph[3] = fminf(fmaxf(hy2[a], 0.f), imh);
    scores[nh] = shv[a] * mw;
  }
}

// -------------------------------------------------- top-k: histogram ----
__global__ void k_hist(const float* __restrict__ scores, unsigned* hist) {
  int i = blockIdx.x * blockDim.x + threadIdx.x;
  if (i >= B_*N_) return;
  int b = i / N_;
  atomicAdd(&hist[b*1024 + (fkey(scores[i]) >> 22)], 1u);
}

// ------------------------------------------- top-k: bucket threshold ----
__global__ void k_thresh(const unsigned* __restrict__ hist, int* cnts) {
  int b = threadIdx.x;
  if (b >= B_) return;
  unsigned cum = 0; int tb = 0; unsigned ca = 0;
  for (int bk = 1023; bk >= 0; --bk) {
    unsigned h = hist[b*1024 + bk];
    if (cum + h >= PRE_) { tb = bk; ca = cum; break; }
    cum += h;
  }
  cnts[b]        = tb;       // threshold bucket
  cnts[B_ + b]   = (int)ca;  // count strictly above
  cnts[2*B_ + b] = 0;        // hi counter
  cnts[3*B_ + b] = 0;        // eq counter
}

// ------------------------------------------------ top-k: compaction ----
__global__ void k_compact(const float* __restrict__ scores, int* cnts,
                          float* tks, int* tki) {
  int i = blockIdx.x * blockDim.x + threadIdx.x;
  if (i >= B_*N_) return;
  int b = i / N_, n = i - b*N_;
  float s = scores[i];
  int bk = (int)(fkey(s) >> 22);
  int tb = cnts[b];
  if (bk > tb) {
    int p = atomicAdd(&cnts[2*B_ + b], 1);
    tks[b*PAD_ + p] = s; tki[b*PAD_ + p] = n;
  } else if (bk == tb) {
    int p = cnts[B_ + b] + atomicAdd(&cnts[3*B_ + b], 1);
    if (p < PRE_) { tks[b*PAD_ + p] = s; tki[b*PAD_ + p] = n; }
  }
}

// --------------------------------- top-k: bitonic sort + box gather ----
__global__ __launch_bounds__(1024) void k_sort(const float* __restrict__ boxes,
                                               float* tks, int* tki, float* tkb) {
  __shared__ float ls[PAD_];
  __shared__ int   li[PAD_];
  int b = blockIdx.x, t = threadIdx.x;
  for (int r = t; r < PAD_; r += 1024) { ls[r] = tks[b*PAD_+r]; li[r] = tki[b*PAD_+r]; }
  __syncthreads();
  for (unsigned kk = 2; kk <= PAD_; kk <<= 1) {
    for (unsigned j = kk >> 1; j > 0; j >>= 1) {
      for (unsigned r = t; r < PAD_; r += 1024) {
        unsigned x = r ^ j;
        if (x > r) {
          bool dir = ((r & kk) == 0);              // true -> descending block
          float s1 = ls[r], s2 = ls[x];
          if ((s1 < s2) == dir) {
            ls[r] = s2; ls[x] = s1;
            int i1 = li[r]; li[r] = li[x]; li[x] = i1;
          }
        }
      }
      __syncthreads();
    }
  }
  for (int r = t; r < PAD_; r += 1024) {
    tks[b*PAD_+r] = ls[r]; tki[b*PAD_+r] = li[r];
    int id = li[r];
    float x1=0.f, y1=0.f, x2=0.f, y2=0.f;
    if (id >= 0) {
      const float* p = boxes + ((size_t)b*N_ + (size_t)id)*4;
      x1 = p[0]; y1 = p[1]; x2 = p[2]; y2 = p[3];
    }
    float* q = tkb + ((size_t)b*PAD_ + r)*4;
    q[0]=x1; q[1]=y1; q[2]=x2; q[3]=y2;
  }
}

// ------------- NMS IoU mask: TDM stage to LDS + WMMA area-sum tiles ----
__global__ __launch_bounds__(128) void k_mask(const float* __restrict__ tkb,
                                              unsigned* mask) {
  __shared__ float lb[PAD_*4];                      // 32 KB: all 2048 boxes
  int b  = blockIdx.x / TILES_;
  int ti = blockIdx.x % TILES_;
  unsigned tid  = threadIdx.x;
  unsigned lane = tid & 31u;
  unsigned wv   = tid >> 5;

  // --- Tensor Data Mover: DMA 2048x4 f32 (32 KB) into LDS, wave0 only
  if (tid < 32) {
    unsigned long long ga = (unsigned long long)(const void*)(tkb + (size_t)b*PAD_*4);
    unsigned ldsa = (unsigned)(unsigned long long)(const void*)lb; // low 32b = LDS offset
    u32x4 g0 = { 1u,                                       // count=1, user D#
                 ldsa,                                     // lds_addr
                 (unsigned)ga,                             // global_addr[31:0]
                 (unsigned)((ga >> 32) & 0x01FFFFFFull) | (2u << 30) }; // addr[56:32] | type=2
    i32x8 g1 = { (int)(2u << 16),        // data_size = 4B
                 (int)(8192u << 16),     // tensor_dim0 lo16 = 8192 elements
                 (int)(1u << 16),        // tensor_dim0 hi16=0 | tensor_dim1 lo16 = 1
                 (int)(8192u << 16),     // tensor_dim1 hi16=0 | tile_dim0 = 8192
                 0,                      // tile_dim1 = 0 (1-D), tile_dim2 = 0
                 8192,                   // tensor_dim0_stride lo32
                 (int)(8192u << 16),     // stride0 hi16=0 | tensor_dim1_stride lo16
                 0 };
    i32x4 z4 = { 0, 0, 0, 0 };
    i32x8 z8 = { 0, 0, 0, 0, 0, 0, 0, 0 };
    __builtin_amdgcn_tensor_load_to_lds(g0, g1, z4, z4, z8, 0);   // 6-arg (clang-23) form
    __builtin_amdgcn_s_wait_tensorcnt(0);
  }
  __syncthreads();

  // row tile: rows rbase..rbase+15 ; per-lane row = v + 8*(lane>=16)
  int rbase = ti * 16;
  int hi = (int)((lane >> 4) << 3);                 // 0 or 8
  float rx1[8], ry1[8], rx2[8], ry2[8];
  #pragma unroll
  for (int v = 0; v < 8; ++v) {
    int R = rbase + v + hi;
    rx1[v] = lb[R*4+0]; ry1[v] = lb[R*4+1];
    rx2[v] = lb[R*4+2]; ry2[v] = lb[R*4+3];
  }
  int mA = (int)(lane & 15u);
  float areaR;
  {
    int R = rbase + mA;
    areaR = (lb[R*4+2]-lb[R*4+0]+1.0f) * (lb[R*4+3]-lb[R*4+1]+1.0f);
  }
  // A (16x4 f32): col0 = area_row, col1 = 1, cols 2..3 = 0 (lanes 16..31 hold K=2,3)
  v2f av = { (lane < 16u) ? areaR : 0.0f, (lane < 16u) ? 1.0f : 0.0f };
  v8f c0 = { 0.f,0.f,0.f,0.f,0.f,0.f,0.f,0.f };
  const float t1 = 1.0f + TNMS;

  for (int tj = (int)wv; tj < TILES_; tj += 4) {
    int C = tj*16 + mA;
    float cx1 = lb[C*4+0], cy1 = lb[C*4+1], cx2 = lb[C*4+2], cy2 = lb[C*4+3];
    float ac  = (cx2-cx1+1.0f) * (cy2-cy1+1.0f);
    // B (4x16 f32): row0 = 1s, row2 = area_col (lanes 16..31), rows 1,3 = 0
    v2f bv = { (lane < 16u) ? 1.0f : ac, 0.0f };
    // D[m][n] = area_row[m]; area_col only meets A's zero col2  (one WMMA vs 8 VALU adds/lane)
    v8f d = __builtin_amdgcn_wmma_f32_16x16x4_f32(false, av, false, bv,
                                                  (short)0, c0, false, false);
    unsigned word = (unsigned)(tj >> 1);
    unsigned shft = (unsigned)(tj & 1) * 16u;
    #pragma unroll
    for (int v = 0; v < 8; ++v) {
      float iw = fminf(rx2[v], cx2) - fmaxf(rx1[v], cx1) + 1.0f; iw = fmaxf(iw, 0.0f);
      float ih = fminf(ry2[v], cy2) - fmaxf(ry1[v], cy1) + 1.0f; ih = fmaxf(ih, 0.0f);
      float inter = iw * ih;
      // inter/(sum-inter) > T  <=>  inter*(1+T) > T*sum
      bool pred = inter * t1 > TNMS * d[v];
      unsigned bal = __builtin_amdgcn_ballot_w32(pred);  // [15:0]=row v, [31:16]=row v+8
      if (lane == 0u) {
        int R0 = rbase + v, R1 = R0 + 8;
        atomicOr(&mask[((size_t)b*PAD_ + R0)*MASKW + word], (bal & 0xFFFFu) << shft);
        atomicOr(&mask[((size_t)b*PAD_ + R1)*MASKW + word], (bal >> 16)     << shft);
      }
    }
  }
}

// --------------------------------------------- greedy suppression scan ----
__global__ __launch_bounds__(64) void k_scan(const unsigned* __restrict__ mask,
                                             unsigned* keep) {
  __shared__ unsigned kw[64];
  int b = blockIdx.x, t = threadIdx.x;
  kw[t] = (t < 62) ? 0xFFFFFFFFu : ((t == 62) ? 0x0000FFFFu : 0u);  // bits >= 2000 off
  __syncthreads();
  for (int i = 0; i < PRE_; ++i) {
    unsigned alive = (kw[i >> 5] >> (i & 31)) & 1u;
    __syncthreads();
    if (alive) {
      unsigned m = mask[((size_t)b*PAD_ + i)*MASKW + t];
      int wi = i >> 5;
      if (t < wi) m = 0u;
      else if (t == wi) {
        int s = (i & 31) + 1;
        m = (s >= 32) ? 0u : (m & (0xFFFFFFFFu << s));   // only j > i suppressed
      }
      kw[t] &= ~m;
    }
    __syncthreads();
  }
  keep[b*64 + t] = kw[t];
}

// --------------------------------------------------- output compaction ----
__global__ __launch_bounds__(64) void k_out(const unsigned* __restrict__ keep,
                                            const float* __restrict__ tkb,
                                            float* out) {
  __shared__ unsigned kw[64];
  __shared__ unsigned pre[64];
  int b = blockIdx.x, t = threadIdx.x;
  kw[t] = keep[b*64 + t];
  __syncthreads();
  if (t == 0) {
    unsigned c = 0;
    for (int w = 0; w < 64; ++w) { pre[w] = c; c += __popc(kw[w]); }
  }
  __syncthreads();
  for (int r = t; r < POST_; r += 64)
    out[((size_t)b*POST_ + r)*5 + 0] = (float)b;          // batch idx column (all rows)
  unsigned w = kw[t];
  unsigned base = pre[t];
  while (w) {
    int bit = __ffs(w) - 1;
    w &= w - 1u;
    int j = t*32 + bit;
    unsigned pos = base + (unsigned)__popc(kw[t] & ((bit == 0) ? 0u : ((1u << bit) - 1u)));
    if (pos < POST_) {
      const float* q = tkb + ((size_t)b*PAD_ + j)*4;
      float* o = out + ((size_t)b*POST_ + pos)*5;
      o[1] = q[0]; o[2] = q[1]; o[3] = q[2]; o[4] = q[3];
    }
  }
}

// -------------------------------------------------------------- launch ----
extern "C" void kernel_launch(void* const* d_in, const int* in_sizes, int n_in,
                              void* d_out, int out_size, void* d_ws, size_t ws_size,
                              hipStream_t stream) {
  const float* sw     = (const float*)d_in[0];
  const float* sh     = (const float*)d_in[1];
  const float* dw     = (const float*)d_in[2];
  const float* dh     = (const float*)d_in[3];
  const float* iminfo = (const float*)d_in[4];
  const float* anw    = (const float*)d_in[5];
  const float* anh    = (const float*)d_in[6];
  float* out = (float*)d_out;

  char* ws = (char*)d_ws;
  size_t off = 0;
  float*    boxes  = (float*)(ws + off);    off += (size_t)B_*N_*4*sizeof(float);     // 35.84 MB
  float*    scores = (float*)(ws + off);    off += (size_t)B_*N_*sizeof(float);       //  8.96 MB
  unsigned* hist   = (unsigned*)(ws + off); off += (size_t)B_*1024*sizeof(unsigned);
  int*      cnts   = (int*)(ws + off);      off += 256;
  float*    tks    = (float*)(ws + off);    off += (size_t)B_*PAD_*sizeof(float);
  int*      tki    = (int*)(ws + off);      off += (size_t)B_*PAD_*sizeof(int);
  float*    tkb    = (float*)(ws + off);    off += (size_t)B_*PAD_*4*sizeof(float);
  unsigned* mask   = (unsigned*)(ws + off); off += (size_t)B_*PAD_*MASKW*sizeof(unsigned); // 8 MB
  unsigned* keep   = (unsigned*)(ws + off); off += (size_t)B_*64*sizeof(unsigned);

  int initN = B_*PAD_*MASKW;                                    // largest region
  k_init   <<<(initN + 255)/256, 256, 0, stream>>>(mask, hist, cnts, tks, tki, out);
  k_prop   <<<(B_*K_ + 255)/256, 256, 0, stream>>>(sw, sh, dw, dh, iminfo, anw, anh, boxes, scores);
  k_hist   <<<(B_*N_ + 255)/256, 256, 0, stream>>>(scores, hist);
  k_thresh <<<1, B_, 0, stream>>>(hist, cnts);
  k_compact<<<(B_*N_ + 255)/256, 256, 0, stream>>>(scores, cnts, tks, tki);
  k_sort   <<<B_, 1024, 0, stream>>>(boxes, tks, tki, tkb);
  k_mask   <<<B_*TILES_, 128, 0, stream>>>(tkb, mask);
  k_scan   <<<B_, 64, 0, stream>>>(mask, keep);
  k_out    <<<B_, 64, 0, stream>>>(keep, tkb, out);
}